// HBS_89275190214712
// MI455X (gfx1250) — compile-verified
//
#include <hip/hip_runtime.h>

#define N_CELLS   50000
#define DEG       32
#define D         64
#define NEG_SLOPE 0.2f

typedef float v2f __attribute__((ext_vector_type(2)));
typedef float v8f __attribute__((ext_vector_type(8)));

// -------- Kernel 1: message = x @ W0 via V_WMMA_F32_16X16X4_F32 --------
// 128 threads = 4 waves; block computes a 16x64 tile of message.
// Wave w computes the 16x16 tile at columns [16w, 16w+16).
__global__ __launch_bounds__(128) void gat_gemm_msg(const float* __restrict__ x,
                                                    const float* __restrict__ W0,
                                                    float* __restrict__ msg) {
  const int lane    = threadIdx.x & 31;
  const int wave    = threadIdx.x >> 5;      // N-tile index 0..3
  const int rowTile = blockIdx.x * 16;
  const int mrow    = lane & 15;             // M within tile (A rows)
  const int hi      = lane >> 4;             // lane half selects K pairs
  const int ncol    = wave * 16 + (lane & 15);

  v8f c = {0.f, 0.f, 0.f, 0.f, 0.f, 0.f, 0.f, 0.f};

#pragma unroll
  for (int k = 0; k < D; k += 4) {
    const int kk = k + (hi << 1);            // lanes 0-15: K=k,k+1 ; 16-31: K=k+2,k+3
    v2f a, b;
    a.x = x[(rowTile + mrow) * D + kk];
    a.y = x[(rowTile + mrow) * D + kk + 1];
    b.x = W0[kk * D + ncol];
    b.y = W0[(kk + 1) * D + ncol];
    c = __builtin_amdgcn_wmma_f32_16x16x4_f32(false, a, false, b,
                                              (short)0, c, false, false);
  }

  // C/D layout: VGPR v -> M=v (lanes 0-15) or M=v+8 (lanes 16-31), N=lane&15
#pragma unroll
  for (int v = 0; v < 8; ++v) {
    const int m = rowTile + v + (hi << 3);
    msg[m * D + ncol] = c[v];
  }
}

// -------- Kernel 2: node_src / node_dst = message @ a_src|a_dst --------
__global__ __launch_bounds__(256) void gat_node_scores(const float* __restrict__ msg,
                                                       const float* __restrict__ a0,
                                                       float* __restrict__ ns,
                                                       float* __restrict__ nd) {
  const int n = blockIdx.x * blockDim.x + threadIdx.x;
  if (n >= N_CELLS) return;
  const float* m = msg + (size_t)n * D;
  float s = 0.f, d = 0.f;
#pragma unroll
  for (int i = 0; i < D; ++i) {
    const float v = m[i];
    s += v * a0[i];        // a_src (wave-uniform -> scalar loads)
    d += v * a0[D + i];    // a_dst
  }
  ns[n] = s;
  nd[n] = d;
}

// -------- Kernel 3: per-edge attention + gather-aggregate, one wave per row ----
// lane = edge index within row (DEG == warpSize == 32).
__global__ __launch_bounds__(256) void gat_edge_agg(const float* __restrict__ msg,
                                                    const int* __restrict__ edge_cols,
                                                    const float* __restrict__ nv,
                                                    const float* __restrict__ ns,
                                                    const float* __restrict__ nd,
                                                    float* __restrict__ out) {
  const int lane = threadIdx.x & 31;
  const int row  = blockIdx.x * (blockDim.x >> 5) + (threadIdx.x >> 5);
  if (row >= N_CELLS) return;

  const long e   = (long)row * DEG + lane;
  const int  col = edge_cols[e];
  const float v  = nv[e];

  const float s  = ns[row] + nd[col];
  const float el = s > 0.f ? s : NEG_SLOPE * s;        // leaky_relu

  // wave32 sum-reduce for row_sum
  float rs = el;
#pragma unroll
  for (int off = 16; off > 0; off >>= 1) rs += __shfl_xor(rs, off, 32);

  const float w = v * (el / rs);                       // nv * att

  // out[row][:] = sum_j w_j * message[col_j][:]; each lane owns 2 dims.
  // Constant-lane broadcasts via v_readlane -> wj/cj live in SGPRs, so the
  // message gather becomes an SGPR-base + per-lane-offset global_load_b64.
  float acc0 = 0.f, acc1 = 0.f;
#pragma unroll
  for (int j = 0; j < DEG; ++j) {
    const float wj = __int_as_float(__builtin_amdgcn_readlane(__float_as_int(w), j));
    const int   cj = __builtin_amdgcn_readlane(col, j);
    const float2 mv = *(const float2*)(msg + (long)cj * D + lane * 2);
    acc0 += wj * mv.x;
    acc1 += wj * mv.y;
  }

  float* orow = out + (long)row * D + lane * 2;
  orow[0] = acc0 > 0.f ? acc0 : 0.f;                   // relu
  orow[1] = acc1 > 0.f ? acc1 : 0.f;
}

extern "C" void kernel_launch(void* const* d_in, const int* in_sizes, int n_in,
                              void* d_out, int out_size, void* d_ws, size_t ws_size,
                              hipStream_t stream) {
  const float* x         = (const float*)d_in[0];   // (50000, 64)
  // d_in[1] = edge_rows: structurally repeat(arange(N), 32) -> not needed
  const int*   edge_cols = (const int*)d_in[2];     // (1.6M,)
  const float* nv        = (const float*)d_in[3];   // (1.6M,)
  const float* W0        = (const float*)d_in[4];   // (64, 64)
  const float* a0        = (const float*)d_in[5];   // (128, 1)
  float* out = (float*)d_out;                       // (50000, 64)

  // Workspace: message (50000*64 f32), node_src (50000), node_dst (50000)
  float* msg = (float*)d_ws;
  float* ns  = msg + (size_t)N_CELLS * D;
  float* nd  = ns + N_CELLS;

  gat_gemm_msg<<<N_CELLS / 16, 128, 0, stream>>>(x, W0, msg);
  gat_node_scores<<<(N_CELLS + 255) / 256, 256, 0, stream>>>(msg, a0, ns, nd);
  gat_edge_agg<<<N_CELLS / 8, 256, 0, stream>>>(msg, edge_cols, nv, ns, nd, out);
}